// GCNNet_77403900609162
// MI455X (gfx1250) — compile-verified
//
#include <hip/hip_runtime.h>
#include <math.h>

// ---------------------------------------------------------------------------
// GCNNet forward for MI455X (gfx1250, wave32).
// Dense node GEMMs run on V_WMMA_F32_16X16X4_F32 (fp32 WMMA, one 16x16 output
// tile per wave). Sparse GCN aggregation is an L2-resident fp32 atomic scatter
// (node buffers ~6.4MB << 192MB L2). Edge indices are converted to int32 and
// the symmetric norm dis[src]*dis[dst] is precomputed once to minimize HBM
// traffic across the 4 layers. Self-loops are folded into a dense init.
// ---------------------------------------------------------------------------

typedef __attribute__((ext_vector_type(2))) float v2f;
typedef __attribute__((ext_vector_type(8))) float v8f;

#define H 32
#define PRE 64

__device__ __forceinline__ float elu1(float v) {
  return v > 0.0f ? v : (expf(v) - 1.0f);
}

// ---- degree / normalization -----------------------------------------------

__global__ void k_init_deg(float* deg, int n) {
  int i = blockIdx.x * blockDim.x + threadIdx.x;
  if (i < n) deg[i] = 1.0f;                      // self-loop contribution
}

__global__ void k_deg_edges(const long long* __restrict__ ei, float* deg, int E) {
  int e = blockIdx.x * blockDim.x + threadIdx.x;
  if (e < E) atomicAdd(&deg[(int)ei[(size_t)E + e]], 1.0f);   // dst row
}

__global__ void k_rsqrt(float* deg, int n) {
  int i = blockIdx.x * blockDim.x + threadIdx.x;
  if (i < n) deg[i] = rsqrtf(deg[i]);            // deg >= 1 always
}

__global__ void k_edge_prep(const long long* __restrict__ ei,
                            const float* __restrict__ dis,
                            int* __restrict__ s32, int* __restrict__ d32,
                            float* __restrict__ norm, int E) {
  int e = blockIdx.x * blockDim.x + threadIdx.x;
  if (e >= E) return;
  int s = (int)ei[e];
  int d = (int)ei[(size_t)E + e];
  s32[e] = s;
  d32[e] = d;
  norm[e] = dis[s] * dis[d];
}

// ---- prefeature layer 1 (2 -> 32, ReLU), K=2 is too small for WMMA --------

__global__ void k_pre1(const float* __restrict__ x, const float* __restrict__ w,
                       const float* __restrict__ b, float* __restrict__ t, int n) {
  int tid = blockIdx.x * blockDim.x + threadIdx.x;
  int i = tid >> 5, j = tid & 31;
  if (i >= n) return;
  float v = fmaf(x[2 * (size_t)i], w[j], fmaf(x[2 * (size_t)i + 1], w[32 + j], b[j]));
  t[(size_t)i * 32 + j] = fmaxf(v, 0.0f);
}

// ---- fp32 WMMA GEMM:  C[n,F] = A[n,K] @ W[K,F] (+bias) (+activation) ------
// One wave computes one 16x16 tile of C via V_WMMA_F32_16X16X4_F32.
// A 16x4 layout : lanes 0-15 hold K=(k0,k0+1), lanes 16-31 hold K=(k0+2,k0+3)
// B 4x16 layout : VGPR0/1 = consecutive K rows, col = lane&15, K split by lane half
// C 16x16 layout: VGPR r -> row r (lanes 0-15) / row r+8 (lanes 16-31)
// act: 0 = none, 2 = ELU
__global__ void k_wmma_gemm(const float* __restrict__ A, const float* __restrict__ W,
                            const float* __restrict__ bias, float* __restrict__ C,
                            int n, int K, int F, int act) {
  int wave = (blockIdx.x * blockDim.x + threadIdx.x) >> 5;
  int lane = threadIdx.x & 31;
  int tilesF = F >> 4;
  int tm = wave / tilesF;
  int tf = wave - tm * tilesF;
  if (tm * 16 >= n) return;                      // wave-uniform exit: EXEC stays full
  int hi   = lane >> 4;                          // 0 or 1
  int m    = tm * 16 + (lane & 15);
  if (m >= n) m = n - 1;                         // safe clamp (n%16==0 in practice)
  int col  = tf * 16 + (lane & 15);
  int kb   = hi * 2;

  v8f acc = {0.f, 0.f, 0.f, 0.f, 0.f, 0.f, 0.f, 0.f};
  const float* arow = A + (size_t)m * K;
  for (int k0 = 0; k0 < K; k0 += 4) {
    v2f a, b;
    a.x = arow[k0 + kb];
    a.y = arow[k0 + kb + 1];
    b.x = W[(size_t)(k0 + kb) * F + col];
    b.y = W[(size_t)(k0 + kb + 1) * F + col];
    acc = __builtin_amdgcn_wmma_f32_16x16x4_f32(
        /*neg_a=*/false, a, /*neg_b=*/false, b,
        /*c_mod=*/(short)0, acc, /*reuse_a=*/false, /*reuse_b=*/false);
  }

  float bv = bias ? bias[col] : 0.0f;
  int rbase = tm * 16 + hi * 8;
#pragma unroll
  for (int r = 0; r < 8; ++r) {
    int row = rbase + r;
    if (row < n) {
      float v = acc[r] + bv;
      if (act == 2) v = elu1(v);
      C[(size_t)row * F + col] = v;
    }
  }
}

// ---- GCN aggregation -------------------------------------------------------

// self-loop: acc[i][c] = hw[i][c] * dis[i]^2
__global__ void k_self_init(const float* __restrict__ hw, const float* __restrict__ dis,
                            float* __restrict__ acc, int n) {
  int tid = blockIdx.x * blockDim.x + threadIdx.x;
  int i = tid >> 5, c = tid & 31;
  if (i >= n) return;
  float d = dis[i];
  acc[(size_t)i * 32 + c] = hw[(size_t)i * 32 + c] * d * d;
}

// 8 threads per edge, float4 gather + 4 fp32 atomic adds (L2-resident)
__global__ void k_scatter(const int* __restrict__ s32, const int* __restrict__ d32,
                          const float* __restrict__ norm,
                          const float* __restrict__ hw, float* __restrict__ acc, int E) {
  long long tid = (long long)blockIdx.x * blockDim.x + threadIdx.x;
  int e = (int)(tid >> 3);
  if (e >= E) return;
  int q = (int)(tid & 7);
  int s = s32[e], d = d32[e];
  float w = norm[e];
  const float4 v = *reinterpret_cast<const float4*>(hw + (size_t)s * 32 + q * 4);
  float* ap = acc + (size_t)d * 32 + q * 4;
  atomicAdd(ap + 0, v.x * w);
  atomicAdd(ap + 1, v.y * w);
  atomicAdd(ap + 2, v.z * w);
  atomicAdd(ap + 3, v.w * w);
}

__global__ void k_finish_elu(const float* __restrict__ acc, const float* __restrict__ b,
                             float* __restrict__ ha, int n) {
  int tid = blockIdx.x * blockDim.x + threadIdx.x;
  int i = tid >> 5, c = tid & 31;
  if (i >= n) return;
  ha[(size_t)i * 32 + c] = elu1(acc[(size_t)i * 32 + c] + b[c]);
}

// ---- heads -----------------------------------------------------------------

__global__ void k_proba(const float* __restrict__ p1, const float* __restrict__ pw2,
                        const float* __restrict__ pb2, float* __restrict__ out, int n) {
  int i = blockIdx.x * blockDim.x + threadIdx.x;
  if (i >= n) return;
  const float* row = p1 + (size_t)i * 32;
  float v = pb2[0];
#pragma unroll
  for (int j = 0; j < 32; ++j) v = fmaf(row[j], pw2[j], v);
  out[i] = v;
}

__global__ void k_zero32(float* buf) {
  if (blockIdx.x == 0 && threadIdx.x < 32) buf[threadIdx.x] = 0.0f;
}

// strided partial sums -> few atomics per column
__global__ void k_mean(const float* __restrict__ ha, float* __restrict__ meanbuf,
                       int n, float invn) {
  int c = threadIdx.x & 31;
  int slice = (blockIdx.x * blockDim.x + threadIdx.x) >> 5;
  int nslices = (gridDim.x * blockDim.x) >> 5;
  float s = 0.0f;
  for (int i = slice; i < n; i += nslices) s += ha[(size_t)i * 32 + c];
  atomicAdd(&meanbuf[c], s * invn);
}

__global__ void k_value(const float* __restrict__ mean, const float* __restrict__ vw1,
                        const float* __restrict__ vb1, const float* __restrict__ vw2,
                        const float* __restrict__ vb2, float* __restrict__ out_val) {
  if (blockIdx.x != 0 || threadIdx.x != 0) return;
  float v = 0.0f;
  for (int j = 0; j < 32; ++j) {
    float a = vb1[j];
    for (int k = 0; k < 32; ++k) a = fmaf(mean[k], vw1[k * 32 + j], a);
    v = fmaf(elu1(a), vw2[j], v);
  }
  *out_val = v + vb2[0];
}

// ---------------------------------------------------------------------------

static inline int cdiv(long long a, long long b) { return (int)((a + b - 1) / b); }

extern "C" void kernel_launch(void* const* d_in, const int* in_sizes, int n_in,
                              void* d_out, int out_size, void* d_ws, size_t ws_size,
                              hipStream_t stream) {
  const float*     x    = (const float*)d_in[0];
  const long long* ei   = (const long long*)d_in[1];
  const float*     f1w  = (const float*)d_in[2];
  const float*     f1b  = (const float*)d_in[3];
  const float*     f2w  = (const float*)d_in[4];
  const float*     f2b  = (const float*)d_in[5];
  const float*     cw[4] = { (const float*)d_in[6],  (const float*)d_in[8],
                             (const float*)d_in[10], (const float*)d_in[12] };
  const float*     cb[4] = { (const float*)d_in[7],  (const float*)d_in[9],
                             (const float*)d_in[11], (const float*)d_in[13] };
  const float*     pw1  = (const float*)d_in[14];
  const float*     pb1  = (const float*)d_in[15];
  const float*     pw2  = (const float*)d_in[16];
  const float*     pb2  = (const float*)d_in[17];
  const float*     vw1  = (const float*)d_in[18];
  const float*     vb1  = (const float*)d_in[19];
  const float*     vw2  = (const float*)d_in[20];
  const float*     vb2  = (const float*)d_in[21];

  const int n = in_sizes[0] / 2;       // x is [N,2]
  const int E = in_sizes[1] / 2;       // edge_index is [2,E]

  // workspace carve-out (256B aligned slabs)
  char* ws = (char*)d_ws;
  size_t off = 0;
  auto take = [&](size_t bytes) {
    void* p = ws + off;
    off += (bytes + 255) & ~(size_t)255;
    return p;
  };
  float* dis     = (float*)take((size_t)n * 4);          // deg -> dis in place
  int*   s32     = (int*)  take((size_t)E * 4);
  int*   d32     = (int*)  take((size_t)E * 4);
  float* norm    = (float*)take((size_t)E * 4);
  float* h0      = (float*)take((size_t)n * PRE * 4);    // prefeature out; later p1
  float* ha      = (float*)take((size_t)n * H * 4);      // layer activations (also pre1 tmp)
  float* hw      = (float*)take((size_t)n * H * 4);      // h @ W
  float* acc     = (float*)take((size_t)n * H * 4);      // scatter accumulator
  float* meanbuf = (float*)take(32 * 4);
  (void)ws_size;

  const int TB = 256;
  float* out = (float*)d_out;          // [n] proba then [1] value

  // 1) degree + normalization
  k_init_deg <<<cdiv(n, TB), TB, 0, stream>>>(dis, n);
  k_deg_edges<<<cdiv(E, TB), TB, 0, stream>>>(ei, dis, E);
  k_rsqrt    <<<cdiv(n, TB), TB, 0, stream>>>(dis, n);
  k_edge_prep<<<cdiv(E, TB), TB, 0, stream>>>(ei, dis, s32, d32, norm, E);

  // 2) prefeature: relu(x@f1w+b) stored in ha[N,32], then WMMA 32->64 into h0
  k_pre1<<<cdiv((long long)n * 32, TB), TB, 0, stream>>>(x, f1w, f1b, ha, n);
  {
    long long waves = (long long)cdiv(n, 16) * (PRE / 16);
    k_wmma_gemm<<<cdiv(waves * 32, TB), TB, 0, stream>>>(ha, f2w, f2b, h0,
                                                         n, 32, PRE, /*act=*/0);
  }

  // 3) four GCNConv + ELU layers
  for (int l = 0; l < 4; ++l) {
    const float* hin = (l == 0) ? h0 : ha;
    const int    K   = (l == 0) ? PRE : H;
    long long waves = (long long)cdiv(n, 16) * (H / 16);
    k_wmma_gemm<<<cdiv(waves * 32, TB), TB, 0, stream>>>(hin, cw[l], nullptr, hw,
                                                         n, K, H, /*act=*/0);
    k_self_init<<<cdiv((long long)n * 32, TB), TB, 0, stream>>>(hw, dis, acc, n);
    k_scatter  <<<cdiv((long long)E * 8,  TB), TB, 0, stream>>>(s32, d32, norm, hw, acc, E);
    k_finish_elu<<<cdiv((long long)n * 32, TB), TB, 0, stream>>>(acc, cb[l], ha, n);
  }

  // 4) policy head: p1 = elu(ha@pw1+pb1) via WMMA (reuse h0), then dot with pw2
  {
    long long waves = (long long)cdiv(n, 16) * (H / 16);
    k_wmma_gemm<<<cdiv(waves * 32, TB), TB, 0, stream>>>(ha, pw1, pb1, h0,
                                                         n, H, H, /*act=*/2);
  }
  k_proba<<<cdiv(n, TB), TB, 0, stream>>>(h0, pw2, pb2, out, n);

  // 5) value head: mean pool + tiny MLP
  k_zero32<<<1, 32, 0, stream>>>(meanbuf);
  k_mean  <<<256, TB, 0, stream>>>(ha, meanbuf, n, 1.0f / (float)n);
  k_value <<<1, 32, 0, stream>>>(meanbuf, vw1, vb1, vw2, vb2, out + n);
}